// RankingLoss_64742337020663
// MI455X (gfx1250) — compile-verified
//
#include <hip/hip_runtime.h>

// Problem constants from the reference
#define NB          32
#define HW          (512 * 1024)      // C=1, H=512, W=1024
#define SAMPLE_NUM  5000
#define TOTAL       (NB * SAMPLE_NUM) // 160000 samples
#define SIGMA       0.03f
#define EPSF        1e-6f
#define ALPHA       1.0f
#define LOSS_WEIGHT 1.0f

#define BLOCKS  160
#define THREADS 256                    // 8 wave32 per block
#define WAVES   (THREADS / 32)

typedef __attribute__((ext_vector_type(2))) float v2f;
typedef __attribute__((ext_vector_type(8))) float v8f;

// Full-precision 32-lane sum using the CDNA5 f32 matrix core.
// A (16x4): lane L<16 holds A[L,0]=a.x, A[L,1]=a.y; lane L+16 holds A[L,2],A[L,3].
// With a.x = x, a.y = 0 and B = ones(4x16): D[m,n] = x(m) + x(m+16) (rowsum_m).
// D layout: VGPR j, lanes 0-15 -> D[j,n] (M=j), lanes 16-31 -> M=j+8.
// Per-lane sum of d[0..7] gives sum(rows 0-7) in lanes 0-15 and sum(rows 8-15)
// in lanes 16-31; one shfl_xor(16) yields the full 32-lane total in every lane.
__device__ __forceinline__ float wave_sum_wmma(float x) {
    v2f a; a.x = x;    a.y = 0.0f;
    v2f b; b.x = 1.0f; b.y = 1.0f;   // B = all ones
    v8f c = {};
    v8f d = __builtin_amdgcn_wmma_f32_16x16x4_f32(
        /*neg_a=*/false, a, /*neg_b=*/false, b,
        /*c_mod=*/(short)0, c, /*reuse_a=*/false, /*reuse_b=*/false);
    float s = d[0] + d[1] + d[2] + d[3] + d[4] + d[5] + d[6] + d[7];
    s += __shfl_xor(s, 16, 32);
    return s;
}

__global__ __launch_bounds__(THREADS)
void rank_reduce_kernel(const float* __restrict__ pred,
                        const float* __restrict__ targ,
                        const unsigned char* __restrict__ mask,
                        const int* __restrict__ idxA,
                        const int* __restrict__ idxB,
                        float* __restrict__ partials /* BLOCKS*3 floats */) {
    const float HI = 1.0f + SIGMA;
    const float LO = 1.0f / (1.0f + SIGMA);

    float eq_sum = 0.0f, un_sum = 0.0f, valid = 0.0f;

    const int stride = gridDim.x * blockDim.x;
    for (int g = blockIdx.x * blockDim.x + threadIdx.x; g < TOTAL; g += stride) {
        const int n = g / SAMPLE_NUM;
        const int s = g - n * SAMPLE_NUM;
        const long long base = (long long)n * HW;
        const int si = n * SAMPLE_NUM + s;

        const int ia = idxA[si];
        const int ib = idxB[si];

        const float pA = pred[base + ia];
        const float pB = pred[base + ib];
        const float tA = targ[base + ia];
        const float tB = targ[base + ib];
        const bool  mA = mask[base + ia] != 0;
        const bool  mB = mask[base + ib] != 0;

        const float ratio  = tA / (tB + EPSF);
        const bool  inband = (ratio < HI) & (ratio > LO);
        const float label  = (ratio >= HI) ? 1.0f : ((ratio <= LO) ? -1.0f : 0.0f);
        const bool  cm     = mA & mB;

        const float diff = pA - pB;
        const float eq_t = diff * diff;
        const float un_t = log1pf(__expf((pB - pA) * label));

        valid  += cm ? 1.0f : 0.0f;
        eq_sum += (cm &&  inband) ? eq_t : 0.0f;
        un_sum += (cm && !inband) ? un_t : 0.0f;
    }

    // Per-wave reduction on the matrix core (EXEC is all-1s here: no thread
    // exited early, the bounds check lives inside the loop above).
    const float weq = wave_sum_wmma(eq_sum);
    const float wun = wave_sum_wmma(un_sum);
    const float wvl = wave_sum_wmma(valid);

    __shared__ float lds[WAVES][3];
    const int wid  = threadIdx.x >> 5;
    const int lane = threadIdx.x & 31;
    if (lane == 0) {
        lds[wid][0] = weq;
        lds[wid][1] = wun;
        lds[wid][2] = wvl;
    }
    __syncthreads();

    if (threadIdx.x == 0) {
        float a = 0.0f, b = 0.0f, c = 0.0f;
        for (int w = 0; w < WAVES; ++w) {
            a += lds[w][0]; b += lds[w][1]; c += lds[w][2];
        }
        partials[blockIdx.x * 3 + 0] = a;
        partials[blockIdx.x * 3 + 1] = b;
        partials[blockIdx.x * 3 + 2] = c;
    }
}

// Deterministic final combine: fixed-order serial sum over per-block partials.
__global__ void rank_finalize_kernel(const float* __restrict__ partials,
                                     float* __restrict__ out) {
    if (blockIdx.x == 0 && threadIdx.x == 0) {
        float eq = 0.0f, un = 0.0f, vl = 0.0f;
        for (int i = 0; i < BLOCKS; ++i) {
            eq += partials[3 * i + 0];
            un += partials[3 * i + 1];
            vl += partials[3 * i + 2];
        }
        out[0] = (ALPHA * eq + un) / (vl + EPSF) * LOSS_WEIGHT;
    }
}

extern "C" void kernel_launch(void* const* d_in, const int* in_sizes, int n_in,
                              void* d_out, int out_size, void* d_ws, size_t ws_size,
                              hipStream_t stream) {
    // setup_inputs() order: prediction, target, mask, idx_A, idx_B
    const float*         pred = (const float*)d_in[0];
    const float*         targ = (const float*)d_in[1];
    const unsigned char* msk  = (const unsigned char*)d_in[2]; // numpy bool: 1B/elem
    const int*           idxA = (const int*)d_in[3];
    const int*           idxB = (const int*)d_in[4];

    float* partials = (float*)d_ws; // BLOCKS*3 floats, fully rewritten every call

    rank_reduce_kernel<<<BLOCKS, THREADS, 0, stream>>>(pred, targ, msk, idxA, idxB,
                                                       partials);
    rank_finalize_kernel<<<1, 32, 0, stream>>>(partials, (float*)d_out);
}